// Radar7PillarNeXtFeatureNet_41721312313898
// MI455X (gfx1250) — compile-verified
//
#include <hip/hip_runtime.h>

// ---------------- problem constants (from reference) ----------------
#define GXD   352
#define GYD   400
#define NBAT  4
#define PPTS  65536
#define NPTS  (NBAT * PPTS)        // 262144
#define GXY   (GXD * GYD)          // 140800
#define NSEG  (NBAT * GXY + 1)     // 563201
#define SENT  (NBAT * GXY)         // 563200 (sentinel segment id)
#define OUTCH 64
#define NTILES (NPTS / 16)         // 16384 WMMA row-tiles
#define SCAN_NB ((NSEG + 1023) / 1024)  // 551

typedef float v2f __attribute__((ext_vector_type(2)));
typedef float v8f __attribute__((ext_vector_type(8)));

// ---------------- shared voxel-coordinate function (division, matches reference) -------
__device__ __forceinline__ void voxel_coords(float px, float py, float pz,
                                             int* cx, int* cy, int* cz, int* seg, int b)
{
    const float VXx = 0.2f, VYy = 0.2f, VZz = 4.0f;
    const float X0 = 0.0f, Y0 = -40.0f, Z0 = -3.0f;
    float cxf = (px - X0) / VXx;           // identical formula in EVERY pass
    float cyf = (py - Y0) / VYy;
    float czf = (pz - Z0) / VZz;
    *cx = (int)cxf; *cy = (int)cyf; *cz = (int)czf;   // trunc == astype(int32)
    bool ok = (cxf >= 0.f) & (cxf < (float)GXD) & (cyf >= 0.f) & (cyf < (float)GYD);
    *seg = ok ? (b * GXY + (*cx) * GYD + (*cy)) : SENT;
}

// ---------------- per-point 13-feature row (padded to 16) ----------------
__device__ __forceinline__ void point_features(
    const float* __restrict__ pts, int i,
    const unsigned* __restrict__ cnt,
    const float* __restrict__ sx, const float* __restrict__ sy,
    const float* __restrict__ sz,
    float f[16], int* segOut)
{
    const float VXx = 0.2f, VYy = 0.2f, VZz = 4.0f;
    const float X0 = 0.0f, Y0 = -40.0f, Z0 = -3.0f;
    const float* p = pts + (size_t)i * 7;
    float px = p[0], py = p[1], pz = p[2];
    float r0 = p[3], r1 = p[4], r2 = p[5], r3 = p[6];

    int cx, cy, cz, seg;
    voxel_coords(px, py, pz, &cx, &cy, &cz, &seg, i >> 16);   // P = 65536

    float c = (float)max(cnt[seg], 1u);
    float mx = sx[seg] / c, my = sy[seg] / c, mz = sz[seg] / c;

    f[0] = px; f[1] = py; f[2] = pz; f[3] = r0; f[4] = r1; f[5] = r2; f[6] = r3;
    f[7] = px - mx; f[8] = py - my; f[9] = pz - mz;
    f[10] = px - ((float)cx * VXx + VXx * 0.5f + X0);
    f[11] = py - ((float)cy * VYy + VYy * 0.5f + Y0);
    f[12] = pz - ((float)cz * VZz + VZz * 0.5f + Z0);
    f[13] = 0.f; f[14] = 0.f; f[15] = 0.f;
    *segOut = seg;
}

// ---------------- utility: zero with 128-bit stores (n in uint4 units) ----------------
__global__ void k_zero128(uint4* __restrict__ p, long n)
{
    long i = (long)blockIdx.x * blockDim.x + threadIdx.x;
    long stride = (long)gridDim.x * blockDim.x;
    uint4 z = {0u, 0u, 0u, 0u};
    for (; i < n; i += stride) p[i] = z;
}

// ---------------- pass 0: bin points, accumulate counts & xyz sums ----------------
__global__ void k_bin(const float* __restrict__ pts,
                      unsigned* __restrict__ cnt,
                      float* __restrict__ sx, float* __restrict__ sy,
                      float* __restrict__ sz)
{
    int i = blockIdx.x * blockDim.x + threadIdx.x;
    if (i >= NPTS) return;
    const float* p = pts + (size_t)i * 7;
    float px = p[0], py = p[1], pz = p[2];
    int cx, cy, cz, seg;
    voxel_coords(px, py, pz, &cx, &cy, &cz, &seg, i >> 16);   // SAME math as WMMA passes
    atomicAdd(&cnt[seg], 1u);
    atomicAdd(&sx[seg], px);
    atomicAdd(&sy[seg], py);
    atomicAdd(&sz[seg], pz);
}

// ---------------- occupancy scan (reproduce sorted-unique compaction) ----------------
__global__ void k_scan1(const unsigned* __restrict__ cnt, unsigned* __restrict__ partials)
{
    __shared__ unsigned red[256];
    int tid = threadIdx.x;
    int base = blockIdx.x * 1024 + tid * 4;
    unsigned s = 0;
#pragma unroll
    for (int i = 0; i < 4; ++i) {
        int d = base + i;
        s += (d < NSEG && cnt[d] > 0u) ? 1u : 0u;
    }
    red[tid] = s;
    __syncthreads();
    for (int off = 128; off > 0; off >>= 1) {
        if (tid < off) red[tid] += red[tid + off];
        __syncthreads();
    }
    if (tid == 0) partials[blockIdx.x] = red[0];
}

__global__ void k_scan2(unsigned* __restrict__ partials, unsigned* __restrict__ U)
{
    if (threadIdx.x == 0 && blockIdx.x == 0) {
        unsigned run = 0;
        for (int i = 0; i < SCAN_NB; ++i) { unsigned v = partials[i]; partials[i] = run; run += v; }
        U[0] = run;
    }
}

__global__ void k_scan3(const unsigned* __restrict__ cnt,
                        const unsigned* __restrict__ partials,
                        unsigned* __restrict__ cidx, unsigned* __restrict__ unq)
{
    __shared__ unsigned ts[256];
    int tid = threadIdx.x;
    int base = blockIdx.x * 1024 + tid * 4;
    unsigned occ[4]; unsigned s = 0;
#pragma unroll
    for (int i = 0; i < 4; ++i) {
        int d = base + i;
        occ[i] = (d < NSEG && cnt[d] > 0u) ? 1u : 0u;
        s += occ[i];
    }
    ts[tid] = s;
    __syncthreads();
    for (int off = 1; off < 256; off <<= 1) {
        unsigned v = (tid >= off) ? ts[tid - off] : 0u;
        __syncthreads();
        ts[tid] += v;
        __syncthreads();
    }
    unsigned run = ts[tid] - s + partials[blockIdx.x];   // exclusive prefix
#pragma unroll
    for (int i = 0; i < 4; ++i) {
        int d = base + i;
        if (d < NSEG) {
            cidx[d] = run;
            if (occ[i]) { unq[run] = (unsigned)d; run++; }
        }
    }
}

// ---------------- pass A: fused feature+GEMM (WMMA f32 16x16x4) + BN stats ----------------
__global__ __launch_bounds__(256) void k_gemm_stats(
    const float* __restrict__ pts, const float* __restrict__ W,
    const unsigned* __restrict__ cnt, const float* __restrict__ sx,
    const float* __restrict__ sy, const float* __restrict__ sz,
    float* __restrict__ statSum, float* __restrict__ statSq)
{
    __shared__ float lsum[OUTCH], lsq[OUTCH];
    for (int t = threadIdx.x; t < OUTCH; t += blockDim.x) { lsum[t] = 0.f; lsq[t] = 0.f; }
    __syncthreads();

    const int lane = threadIdx.x & 31;
    const int w = threadIdx.x >> 5;
    const int hv = (lane >> 4) << 1;     // 0 (lanes 0-15: K0,K1) or 2 (lanes 16-31: K2,K3)
    const int nl = lane & 15;
    const int tstride = gridDim.x * 8;

    // B fragments: B[k][n] = W[n][k], k padded past 13 with zeros. Loop-invariant.
    v2f bfr[4][4];
#pragma unroll
    for (int g = 0; g < 4; ++g) {
        int n = g * 16 + nl;
#pragma unroll
        for (int c = 0; c < 4; ++c) {
            int k = c * 4 + hv;
            bfr[g][c].x = (k     < 13) ? W[n * 13 + k]     : 0.f;
            bfr[g][c].y = (k + 1 < 13) ? W[n * 13 + k + 1] : 0.f;
        }
    }

    float sAcc[4] = {0.f, 0.f, 0.f, 0.f};
    float qAcc[4] = {0.f, 0.f, 0.f, 0.f};

    for (int t = blockIdx.x * 8 + w; t < NTILES; t += tstride) {
        int i = t * 16 + nl;             // both wave halves compute the same 16 rows
        __builtin_prefetch(pts + (size_t)(i + tstride * 16) * 7, 0, 1);  // global_prefetch_b8
        float f[16]; int seg;
        point_features(pts, i, cnt, sx, sy, sz, f, &seg);
        v2f a[4];
#pragma unroll
        for (int c = 0; c < 4; ++c) {    // A fragment: row m=lane%16, K-pair selected by half
            a[c].x = hv ? f[c * 4 + 2] : f[c * 4 + 0];
            a[c].y = hv ? f[c * 4 + 3] : f[c * 4 + 1];
        }
#pragma unroll
        for (int g = 0; g < 4; ++g) {
            v8f acc = {};
#pragma unroll
            for (int c = 0; c < 4; ++c)
                acc = __builtin_amdgcn_wmma_f32_16x16x4_f32(
                    false, a[c], false, bfr[g][c], (short)0, acc, false, false);
            float s = 0.f, q = 0.f;
#pragma unroll
            for (int j = 0; j < 8; ++j) { float v = acc[j]; s += v; q += v * v; }
            sAcc[g] += s; qAcc[g] += q;
        }
    }
#pragma unroll
    for (int g = 0; g < 4; ++g) {        // fold the two wave halves (same channel set)
        sAcc[g] += __shfl_xor(sAcc[g], 16, 32);
        qAcc[g] += __shfl_xor(qAcc[g], 16, 32);
    }
    if (lane < 16) {
#pragma unroll
        for (int g = 0; g < 4; ++g) {
            atomicAdd(&lsum[g * 16 + lane], sAcc[g]);
            atomicAdd(&lsq [g * 16 + lane], qAcc[g]);
        }
    }
    __syncthreads();
    for (int t = threadIdx.x; t < OUTCH; t += blockDim.x) {
        atomicAdd(&statSum[t], lsum[t]);
        atomicAdd(&statSq[t],  lsq[t]);
    }
}

// ---------------- BN finalize: a = gamma*rsqrt(var+eps), b = beta - mu*a ----------------
__global__ void k_bnfin(const float* __restrict__ statSum, const float* __restrict__ statSq,
                        const float* __restrict__ gamma, const float* __restrict__ beta,
                        float* __restrict__ bnA, float* __restrict__ bnB)
{
    int t = threadIdx.x;
    if (t < OUTCH) {
        float mu  = statSum[t] / (float)NPTS;
        float var = statSq[t] / (float)NPTS - mu * mu;
        float a = gamma[t] * rsqrtf(var + 1e-3f);
        bnA[t] = a;
        bnB[t] = beta[t] - mu * a;
    }
}

// ---------------- pass B: recompute GEMM, BN+ReLU, scatter-max into compacted pillars ---
__global__ __launch_bounds__(256) void k_gemm_max(
    const float* __restrict__ pts, const float* __restrict__ W,
    const unsigned* __restrict__ cnt, const float* __restrict__ sx,
    const float* __restrict__ sy, const float* __restrict__ sz,
    const unsigned* __restrict__ cidx,
    const float* __restrict__ bnA, const float* __restrict__ bnB,
    float* __restrict__ out)
{
    const int lane = threadIdx.x & 31;
    const int w = threadIdx.x >> 5;
    const int hv = (lane >> 4) << 1;
    const int nl = lane & 15;
    const int tstride = gridDim.x * 8;

    v2f bfr[4][4];
    float An[4], Bn[4];
#pragma unroll
    for (int g = 0; g < 4; ++g) {
        int n = g * 16 + nl;
        An[g] = bnA[n];
        Bn[g] = bnB[n];
#pragma unroll
        for (int c = 0; c < 4; ++c) {
            int k = c * 4 + hv;
            bfr[g][c].x = (k     < 13) ? W[n * 13 + k]     : 0.f;
            bfr[g][c].y = (k + 1 < 13) ? W[n * 13 + k + 1] : 0.f;
        }
    }

    for (int t = blockIdx.x * 8 + w; t < NTILES; t += tstride) {
        int i = t * 16 + nl;
        __builtin_prefetch(pts + (size_t)(i + tstride * 16) * 7, 0, 1);  // global_prefetch_b8
        float f[16]; int seg;
        point_features(pts, i, cnt, sx, sy, sz, f, &seg);
        unsigned cv = (seg == SENT) ? 0xFFFFFFFFu : cidx[seg];  // compacted pillar (or skip)
        v2f a[4];
#pragma unroll
        for (int c = 0; c < 4; ++c) {
            a[c].x = hv ? f[c * 4 + 2] : f[c * 4 + 0];
            a[c].y = hv ? f[c * 4 + 3] : f[c * 4 + 1];
        }
#pragma unroll
        for (int g = 0; g < 4; ++g) {
            v8f acc = {};
#pragma unroll
            for (int c = 0; c < 4; ++c)
                acc = __builtin_amdgcn_wmma_f32_16x16x4_f32(
                    false, a[c], false, bfr[g][c], (short)0, acc, false, false);
#pragma unroll
            for (int j = 0; j < 8; ++j) {
                int mrow = j + ((lane >> 4) << 3);               // D layout: M = j (+8 hi half)
                unsigned cs = (unsigned)__shfl((int)cv, mrow, 32);
                float v = acc[j] * An[g] + Bn[g];
                v = v > 0.f ? v : 0.f;                           // ReLU => v >= 0
                if (cs != 0xFFFFFFFFu)
                    atomicMax((unsigned*)(out + (size_t)cs * OUTCH + (g * 16 + nl)),
                              __float_as_uint(v));               // u32 order == f32 order (>=0)
            }
        }
    }
}

// ---------------- coords + grid_size writeout ----------------
__global__ void k_coords(const unsigned* __restrict__ unq, const unsigned* __restrict__ U,
                         int* __restrict__ coordsOut)
{
    int j = blockIdx.x * blockDim.x + threadIdx.x;
    if (j == 0) {                                // grid_size = [GY, GX]
        coordsOut[(size_t)NSEG * 3 + 0] = GYD;
        coordsOut[(size_t)NSEG * 3 + 1] = GXD;
    }
    if (j < NSEG) {
        unsigned Uv = U[0];
        unsigned d = (j < (int)Uv) ? unq[j] : (unsigned)SENT;    // pad with sentinel
        int b = d / GXY;
        int rem = d % GXY;
        int cx = rem / GYD;
        int cy = rem % GYD;
        coordsOut[(size_t)j * 3 + 0] = b;
        coordsOut[(size_t)j * 3 + 1] = cy;
        coordsOut[(size_t)j * 3 + 2] = cx;
    }
}

// ---------------- launch ----------------
extern "C" void kernel_launch(void* const* d_in, const int* in_sizes, int n_in,
                              void* d_out, int out_size, void* d_ws, size_t ws_size,
                              hipStream_t stream)
{
    const float* pts   = (const float*)d_in[0];
    const float* W     = (const float*)d_in[1];
    const float* gamma = (const float*)d_in[2];
    const float* beta  = (const float*)d_in[3];
    float* out = (float*)d_out;

    // 16-byte-aligned workspace carve-out (so k_zero128 can use b128 stores)
    uintptr_t p = (uintptr_t)d_ws;
    auto carve = [&p](size_t bytes) -> uintptr_t {
        p = (p + 15u) & ~(uintptr_t)15u;
        uintptr_t r = p;
        p += bytes;
        return r;
    };
    unsigned* cnt      = (unsigned*)carve(sizeof(unsigned) * NSEG);
    float*    sx       = (float*)carve(sizeof(float) * NSEG);
    float*    sy       = (float*)carve(sizeof(float) * NSEG);
    float*    sz       = (float*)carve(sizeof(float) * NSEG);
    unsigned* cidx     = (unsigned*)carve(sizeof(unsigned) * NSEG);
    unsigned* unq      = (unsigned*)carve(sizeof(unsigned) * NSEG);
    unsigned* partials = (unsigned*)carve(sizeof(unsigned) * 1024);
    unsigned* U        = (unsigned*)carve(sizeof(unsigned) * 4);
    float*    statSum  = (float*)carve(sizeof(float) * OUTCH * 2);   // statSum|statSq
    float*    statSq   = statSum + OUTCH;
    float*    bnA      = (float*)carve(sizeof(float) * OUTCH);
    float*    bnB      = (float*)carve(sizeof(float) * OUTCH);
    (void)in_sizes; (void)n_in; (void)out_size; (void)ws_size;

    // zero-init with 128-bit stores: bin tables (cnt,sx,sy,sz), BN stats, output feats.
    // NSEG*4 bytes per table is not a multiple of 16 -> round word counts up inside
    // each region (regions are 16B aligned and carved with headroom via alignment).
    k_zero128<<<1024, 256, 0, stream>>>((uint4*)cnt, ((long)NSEG + 3) / 4);
    k_zero128<<<1024, 256, 0, stream>>>((uint4*)sx,  ((long)NSEG + 3) / 4);
    k_zero128<<<1024, 256, 0, stream>>>((uint4*)sy,  ((long)NSEG + 3) / 4);
    k_zero128<<<1024, 256, 0, stream>>>((uint4*)sz,  ((long)NSEG + 3) / 4);
    k_zero128<<<1, 32, 0, stream>>>((uint4*)statSum, (2 * OUTCH) / 4);
    k_zero128<<<4096, 256, 0, stream>>>((uint4*)out, ((long)NSEG * OUTCH) / 4);

    k_bin<<<NPTS / 256, 256, 0, stream>>>(pts, cnt, sx, sy, sz);

    k_scan1<<<SCAN_NB, 256, 0, stream>>>(cnt, partials);
    k_scan2<<<1, 1, 0, stream>>>(partials, U);
    k_scan3<<<SCAN_NB, 256, 0, stream>>>(cnt, partials, cidx, unq);

    k_gemm_stats<<<256, 256, 0, stream>>>(pts, W, cnt, sx, sy, sz, statSum, statSq);
    k_bnfin<<<1, 64, 0, stream>>>(statSum, statSq, gamma, beta, bnA, bnB);
    k_gemm_max<<<256, 256, 0, stream>>>(pts, W, cnt, sx, sy, sz, cidx, bnA, bnB, out);

    k_coords<<<(NSEG + 255) / 256, 256, 0, stream>>>(unq, U, (int*)(out + (size_t)NSEG * OUTCH));
}